// feature_network_37546604101648
// MI455X (gfx1250) — compile-verified
//
#include <hip/hip_runtime.h>
#include <math.h>

// ---------------------------------------------------------------------------
// CDNA5 (gfx1250) implementation of the graph feature network.
// All matmuls use V_WMMA_F32_16X16X4_F32 (full fp32 precision, wave32).
// Predictable global streams are prefetched via global_prefetch_b8.
// ---------------------------------------------------------------------------

typedef __attribute__((ext_vector_type(2))) float v2f;
typedef __attribute__((ext_vector_type(8))) float v8f;

static constexpr int Bn = 8;
static constexpr int Nn = 1024;
static constexpr int Kn = 16;
static constexpr int Tn = 16;
static constexpr int Mn = Bn * Nn;   // 8192 nodes
static constexpr int En = Mn * Kn;   // 131072 edges
#define EPSBN 1e-5f

__device__ __forceinline__ v8f vzero8() {
  v8f z = {0.f, 0.f, 0.f, 0.f, 0.f, 0.f, 0.f, 0.f};
  return z;
}

__device__ __forceinline__ v8f wmma4(v2f a, v2f b, v8f c) {
  // D = A(16x4 f32) * B(4x16 f32) + C(16x16 f32)
  return __builtin_amdgcn_wmma_f32_16x16x4_f32(false, a, false, b, (short)0, c,
                                               false, false);
}

// ---------------------------------------------------------------------------
// Generic tiled GEMM:  Y[M,COUT] = X[M,CIN] * W^T + bias,  W is [COUT,CIN]
// one wave computes a 16x16 output tile; 8 waves (256 thr) per block.
// ---------------------------------------------------------------------------
__global__ void gemm_bias_kernel(const float* __restrict__ X,
                                 const float* __restrict__ W,
                                 const float* __restrict__ bias,
                                 float* __restrict__ Y, int CIN, int COUT) {
  const int wave = threadIdx.x >> 5;
  const int lane = threadIdx.x & 31;
  const int half = lane >> 4;
  const int r = lane & 15;
  const int tiles_col = COUT >> 4;
  const int tile = blockIdx.x * 8 + wave;
  const int tm = tile / tiles_col;
  const int tn = tile - tm * tiles_col;
  const int m0 = tm << 4, n0 = tn << 4;

  const float* xp = X + (size_t)(m0 + r) * CIN + 2 * half;
  const float* wp = W + (size_t)(n0 + r) * CIN + 2 * half;

  v8f acc = vzero8();
  for (int k0 = 0; k0 < CIN; k0 += 4) {
    if ((k0 & 63) == 0) {
      // speculative prefetch of the streams ~1 cacheline block ahead
      __builtin_prefetch(xp + k0 + 64, 0, 0);
      __builtin_prefetch(wp + k0 + 64, 0, 0);
    }
    v2f a = *reinterpret_cast<const v2f*>(xp + k0);
    v2f b = *reinterpret_cast<const v2f*>(wp + k0);
    acc = wmma4(a, b, acc);
  }
  const float bb = bias ? bias[n0 + r] : 0.f;
#pragma unroll
  for (int v = 0; v < 8; ++v)
    Y[(size_t)(m0 + v + 8 * half) * COUT + n0 + r] = acc[v] + bb;
}

// ---------------------------------------------------------------------------
// Per-channel BN stats (train mode): one block per channel.
// ---------------------------------------------------------------------------
__global__ void colstats_kernel(const float* __restrict__ Y, int C,
                                const float* __restrict__ g,
                                const float* __restrict__ be,
                                float* __restrict__ scale,
                                float* __restrict__ shift) {
  __shared__ float s1[256], s2[256];
  const int c = blockIdx.x;
  float a = 0.f, b = 0.f;
  for (int row = threadIdx.x; row < Mn; row += 256) {
    float v = Y[(size_t)row * C + c];
    a += v;
    b += v * v;
  }
  s1[threadIdx.x] = a;
  s2[threadIdx.x] = b;
  __syncthreads();
  for (int s = 128; s > 0; s >>= 1) {
    if (threadIdx.x < s) {
      s1[threadIdx.x] += s1[threadIdx.x + s];
      s2[threadIdx.x] += s2[threadIdx.x + s];
    }
    __syncthreads();
  }
  if (threadIdx.x == 0) {
    const float m = s1[0] / (float)Mn;
    const float var = s2[0] / (float)Mn - m * m;
    const float sc = g[c] * rsqrtf(var + EPSBN);
    scale[c] = sc;
    shift[c] = be[c] - m * sc;
  }
}

// BN apply + optional residual + ReLU (elementwise).
__global__ void bn_apply_kernel(const float* __restrict__ Yin,
                                const float* __restrict__ res,
                                const float* __restrict__ scale,
                                const float* __restrict__ shift,
                                float* __restrict__ Yout, int C, int total) {
  const int i = blockIdx.x * 256 + threadIdx.x;
  if (i >= total) return;
  const int c = i & (C - 1);
  float v = scale[c] * Yin[i] + shift[c];
  if (res) v += res[i];
  Yout[i] = fmaxf(v, 0.f);
}

// ---------------------------------------------------------------------------
// Edge-feature sufficient statistics (for analytic BN folding of etype l1).
// stats[0..4] = Sum e0, Sum e1, Sum e0^2, Sum e0*e1, Sum e1^2
// ---------------------------------------------------------------------------
__global__ void zero_kernel(float* p, int n) {
  const int i = blockIdx.x * 64 + threadIdx.x;
  if (i < n) p[i] = 0.f;
}

__global__ void efstats_kernel(const float* __restrict__ pts,
                               const int* __restrict__ nn,
                               float* __restrict__ stats) {
  __shared__ float red[256];
  const int e = blockIdx.x * 256 + threadIdx.x;
  const int node = e >> 4;
  const int b = node >> 10;
  const int n = node & 1023;
  const int id = nn[e];
  const float* pb = pts + (size_t)b * 2048;
  const float e0 = pb[id] - pb[n];
  const float e1 = pb[1024 + id] - pb[1024 + n];
  const float vals[5] = {e0, e1, e0 * e0, e0 * e1, e1 * e1};
  for (int j = 0; j < 5; ++j) {
    red[threadIdx.x] = vals[j];
    __syncthreads();
    for (int s = 128; s > 0; s >>= 1) {
      if (threadIdx.x < s) red[threadIdx.x] += red[threadIdx.x + s];
      __syncthreads();
    }
    if (threadIdx.x == 0) atomicAdd(&stats[j], red[0]);
    __syncthreads();
  }
}

// Fold BN into etype l1 weights: effW = s*W, effB = s*b + be - s*mu.
__global__ void fold_l1_kernel(const float* __restrict__ stats,
                               const float* __restrict__ w,   // [64,2]
                               const float* __restrict__ b,
                               const float* __restrict__ g,
                               const float* __restrict__ be,
                               float* __restrict__ effW, float* __restrict__ effB) {
  const int c = threadIdx.x;
  if (c >= 64) return;
  const float invE = 1.f / (float)En;
  const float m0 = stats[0] * invE, m1 = stats[1] * invE;
  const float c00 = stats[2] * invE - m0 * m0;
  const float c01 = stats[3] * invE - m0 * m1;
  const float c11 = stats[4] * invE - m1 * m1;
  const float w0 = w[c * 2], w1 = w[c * 2 + 1];
  const float mu = w0 * m0 + w1 * m1 + b[c];
  const float var = w0 * w0 * c00 + 2.f * w0 * w1 * c01 + w1 * w1 * c11;
  const float s = g[c] * rsqrtf(var + EPSBN);
  effW[c * 2] = s * w0;
  effW[c * 2 + 1] = s * w1;
  effB[c] = s * b[c] + be[c] - s * mu;
}

// Fully-fused etype network: per edge, relu(l1_folded) -> l2 -> tanh.
__global__ void etype_forward_kernel(const float* __restrict__ pts,
                                     const int* __restrict__ nn,
                                     const float* __restrict__ effW,
                                     const float* __restrict__ effB,
                                     const float* __restrict__ W2,  // [16,64]
                                     const float* __restrict__ b2,  // [16]
                                     float* __restrict__ etype) {
  __shared__ float sW[128], sB[64], sW2[1024], sb2[16];
  const int tid = threadIdx.x;
  if (tid < 128) sW[tid] = effW[tid];
  if (tid < 64) sB[tid] = effB[tid];
  if (tid < 16) sb2[tid] = b2[tid];
  for (int j = tid; j < 1024; j += 256) sW2[j] = W2[j];
  __syncthreads();

  const int e = blockIdx.x * 256 + tid;
  const int node = e >> 4;
  const int b = node >> 10;
  const int n = node & 1023;
  const int id = nn[e];
  const float* pb = pts + (size_t)b * 2048;
  const float e0 = pb[id] - pb[n];
  const float e1 = pb[1024 + id] - pb[1024 + n];

  float h[64];
#pragma unroll
  for (int c = 0; c < 64; ++c)
    h[c] = fmaxf(sW[2 * c] * e0 + sW[2 * c + 1] * e1 + sB[c], 0.f);

  float* outp = etype + (size_t)e * Tn;
#pragma unroll
  for (int t = 0; t < 16; ++t) {
    float s = sb2[t];
    const float* w2 = sW2 + t * 64;
#pragma unroll
    for (int c = 0; c < 64; ++c) s += w2[c] * h[c];
    outp[t] = tanhf(s);
  }
}

// pts [B,2,N] channel-major -> node-major [M,2]
__global__ void pts_nm_kernel(const float* __restrict__ pts, float* __restrict__ x0) {
  const int node = blockIdx.x * 256 + threadIdx.x;
  if (node >= Mn) return;
  const int b = node >> 10, n = node & 1023;
  x0[node * 2] = pts[(size_t)b * 2048 + n];
  x0[node * 2 + 1] = pts[(size_t)b * 2048 + 1024 + n];
}

// ---------------------------------------------------------------------------
// Message-passing conv:  per 16-node tile,
//   agg[i][t][c] = (1/K) sum_k etype[i,k,t] * X[nbr(i,k)][c]   (VALU + LDS)
//   msg[i][d]   += agg[i][:,:] * wt[:,:,d]                     (WMMA GEMM)
// C-dimension is processed in chunks of Cc (kept fully in 320KB LDS).
// ---------------------------------------------------------------------------
template <int CIN, int DOUT>
__global__ void mp_kernel(const float* __restrict__ X, const int* __restrict__ nn,
                          const float* __restrict__ et_g,
                          const float* __restrict__ wt,   // [T, CIN, DOUT]
                          const float* __restrict__ bias, // [DOUT]
                          float* __restrict__ Y) {
  constexpr int Cc = (CIN < 64) ? CIN : 64;
  constexpr int NCH = CIN / Cc;
  constexpr int KK = Tn * Cc;         // contraction length per chunk
  constexpr int TILES = DOUT / 16;
  constexpr int TPW = (TILES + 7) / 8;

  extern __shared__ float smem[];
  float* et = smem;                   // [16][16][16] = 4096
  float* xn = et + 4096;              // [16 nodes][16 k][Cc]
  float* agg = xn + 256 * Cc;         // [16 nodes][16 t][Cc]
  int* gidx = (int*)(agg + 256 * Cc); // [256]

  const int tid = threadIdx.x;
  const int wave = tid >> 5, lane = tid & 31;
  const int half = lane >> 4, r = lane & 15;
  const int node0 = blockIdx.x * 16;
  const int bnum = node0 >> 10;

  gidx[tid] = bnum * 1024 + nn[(size_t)node0 * Kn + tid];
  for (int j = tid; j < 4096; j += 256) et[j] = et_g[(size_t)node0 * 256 + j];

  v8f acc[TPW];
  for (int q = 0; q < TPW; ++q) acc[q] = vzero8();

  const int i = tid >> 4;   // node within tile (agg stage)
  const int t = tid & 15;   // edge type       (agg stage)

  for (int ch = 0; ch < NCH; ++ch) {
    __syncthreads();  // protect agg/xn from previous-chunk readers
    {
      const float* src = X + (size_t)gidx[tid] * CIN + ch * Cc;
      __builtin_prefetch(src, 0, 0);
      float* dst = xn + tid * Cc;
      for (int c = 0; c < Cc; ++c) dst[c] = src[c];
    }
    __syncthreads();
    {
      float etr[16];
#pragma unroll
      for (int k = 0; k < 16; ++k) etr[k] = et[(i * 16 + k) * 16 + t];
      float* ap = agg + i * KK + t * Cc;
      const float* xb = xn + i * 16 * Cc;
      for (int c = 0; c < Cc; ++c) {
        float s = 0.f;
#pragma unroll
        for (int k = 0; k < 16; ++k) s += etr[k] * xb[k * Cc + c];
        ap[c] = s * 0.0625f;  // / K
      }
    }
    __syncthreads();
    // WMMA GEMM over this chunk (wave-uniform control flow: EXEC all-ones)
    for (int q = 0; q < TPW; ++q) {
      const int tix = wave + q * 8;
      if (tix < TILES) {
        const int d0 = tix * 16;
        const float* arow = agg + r * KK;
        for (int kk = 0; kk < KK; kk += 4) {
          const int kb = kk + 2 * half;
          v2f a;
          a.x = arow[kb];
          a.y = arow[kb + 1];
          const int t0 = kb / Cc, c0 = kb % Cc;
          const int t1 = (kb + 1) / Cc, c1 = (kb + 1) % Cc;
          if ((kk & 63) == 0)
            __builtin_prefetch(&wt[((size_t)t0 * CIN + ch * Cc + c0 + 16) * DOUT + d0 + r], 0, 0);
          v2f bv;
          bv.x = wt[((size_t)t0 * CIN + ch * Cc + c0) * DOUT + d0 + r];
          bv.y = wt[((size_t)t1 * CIN + ch * Cc + c1) * DOUT + d0 + r];
          acc[q] = wmma4(a, bv, acc[q]);
        }
      }
    }
  }
  for (int q = 0; q < TPW; ++q) {
    const int tix = wave + q * 8;
    if (tix < TILES) {
      const int d0 = tix * 16;
      const float bb = bias[d0 + r];
#pragma unroll
      for (int v = 0; v < 8; ++v)
        Y[(size_t)(node0 + v + 8 * half) * DOUT + d0 + r] = acc[q][v] + bb;
    }
  }
}

// Final L2-normalize over 512 channels, apply mask, write [B,512,N].
__global__ void norm_out_kernel(const float* __restrict__ X,
                                const float* __restrict__ mask,
                                float* __restrict__ out) {
  __shared__ float red[256];
  const int node = blockIdx.x;
  const int b = node >> 10, n = node & 1023;
  const float* xp = X + (size_t)node * 512;
  const float v0 = xp[threadIdx.x];
  const float v1 = xp[threadIdx.x + 256];
  red[threadIdx.x] = v0 * v0 + v1 * v1;
  __syncthreads();
  for (int s = 128; s > 0; s >>= 1) {
    if (threadIdx.x < s) red[threadIdx.x] += red[threadIdx.x + s];
    __syncthreads();
  }
  const float inv = 1.f / fmaxf(sqrtf(red[0]), 1e-12f);
  const float mk = mask[node];
  out[((size_t)b * 512 + threadIdx.x) * 1024 + n] = v0 * inv * mk;
  out[((size_t)b * 512 + threadIdx.x + 256) * 1024 + n] = v1 * inv * mk;
}

// ---------------------------------------------------------------------------
// Host side
// ---------------------------------------------------------------------------
struct ConvP { const float *b, *be, *g, *w; };
struct MpP   { const float *b, *be, *g, *wt; };

static void run_gemm(const float* X, const float* W, const float* bias, float* Y,
                     int CIN, int COUT, hipStream_t s) {
  const int tiles = (Mn / 16) * (COUT / 16);
  gemm_bias_kernel<<<tiles / 8, 256, 0, s>>>(X, W, bias, Y, CIN, COUT);
}

static void run_bn(const float* Y, int C, const float* g, const float* be,
                   const float* res, float* dst, float* scale, float* shift,
                   hipStream_t s) {
  colstats_kernel<<<C, 256, 0, s>>>(Y, C, g, be, scale, shift);
  bn_apply_kernel<<<(Mn * C) / 256, 256, 0, s>>>(Y, res, scale, shift, dst, C, Mn * C);
}

template <int CIN, int DOUT>
static void run_mp(const float* X, const int* nn, const float* et, const MpP& P,
                   float* Y, hipStream_t s) {
  constexpr int Cc = (CIN < 64) ? CIN : 64;
  const size_t sh = (size_t)(4096 + 2 * 256 * Cc) * sizeof(float) + 256 * sizeof(int);
  (void)hipFuncSetAttribute(reinterpret_cast<const void*>(&mp_kernel<CIN, DOUT>),
                            hipFuncAttributeMaxDynamicSharedMemorySize, (int)sh);
  mp_kernel<CIN, DOUT><<<Mn / 16, 256, sh, s>>>(X, nn, et, P.wt, P.b, Y);
}

extern "C" void kernel_launch(void* const* d_in, const int* in_sizes, int n_in,
                              void* d_out, int out_size, void* d_ws, size_t ws_size,
                              hipStream_t stream) {
  (void)in_sizes; (void)n_in; (void)out_size;

  // --- inputs: top level in setup_inputs() insertion order; params pytree
  // flattened with alphabetically-sorted dict keys at every level (JAX order).
  const float* pts = (const float*)d_in[0];
  const int* nn = (const int*)d_in[1];
  const float* mask = (const float*)d_in[2];

  int p = 3;
  auto F = [&](int idx) { return (const float*)d_in[idx]; };
  auto conv = [&]() { ConvP c; c.b = F(p++); c.be = F(p++); c.g = F(p++); c.w = F(p++); return c; };
  auto mpp  = [&]() { MpP m; m.b = F(p++); m.be = F(p++); m.g = F(p++); m.wt = F(p++); return m; };

  const float* aff_w = F(p++);                 // affine: w only
  ConvP c1 = conv(), c2 = conv(), c3 = conv(), c4 = conv(), c5 = conv();
  // etype.l1 (b,be,g,w), etype.l2 (b,w)
  const float* el1_b = F(p++); const float* el1_be = F(p++);
  const float* el1_g = F(p++); const float* el1_w = F(p++);
  const float* el2_b = F(p++); const float* el2_w = F(p++);
  MpP mp1 = mpp(), mp2 = mpp(), mp3 = mpp();
  ConvP r1d = conv(); MpP r1m = mpp(); ConvP r1u = conv();
  ConvP r2d = conv(); MpP r2m = mpp(); ConvP r2u = conv();
  ConvP r3d = conv(); MpP r3m = mpp(); ConvP r3u = conv();

  // --- workspace layout (floats)
  float* ws = (float*)d_ws;
  float* estats = ws;                                // 8
  float* effW = ws + 8;                              // 128
  float* effB = ws + 136;                            // 64
  float* scaleB = ws + 256;                          // <=512
  float* shiftB = ws + 768;                          // <=512
  float* bufA = ws + 1280;                           // M*512
  float* bufB = bufA + (size_t)Mn * 512;             // M*512
  float* bufC = bufB + (size_t)Mn * 512;             // M*512
  float* etyp = bufC + (size_t)Mn * 512;             // E*16
  float* x0 = etyp + (size_t)En * Tn;                // M*2
  const size_t need = (size_t)(x0 - ws) + (size_t)Mn * 2;
  if (ws_size < need * sizeof(float)) return;

  // --- etype network (BN folded analytically) ---
  zero_kernel<<<1, 64, 0, stream>>>(estats, 8);
  efstats_kernel<<<En / 256, 256, 0, stream>>>(pts, nn, estats);
  fold_l1_kernel<<<1, 64, 0, stream>>>(estats, el1_w, el1_b, el1_g, el1_be, effW, effB);
  etype_forward_kernel<<<En / 256, 256, 0, stream>>>(pts, nn, effW, effB, el2_w, el2_b, etyp);
  pts_nm_kernel<<<Mn / 256, 256, 0, stream>>>(pts, x0);

  // --- mp1 + conv1 ---
  run_mp<2, 64>(x0, nn, etyp, mp1, bufA, stream);
  run_bn(bufA, 64, mp1.g, mp1.be, nullptr, bufA, scaleB, shiftB, stream);
  run_gemm(bufA, c1.w, c1.b, bufB, 64, 128, stream);
  run_bn(bufB, 128, c1.g, c1.be, nullptr, bufB, scaleB, shiftB, stream);

  // --- res1 ---
  run_gemm(bufB, r1d.w, r1d.b, bufA, 128, 64, stream);
  run_bn(bufA, 64, r1d.g, r1d.be, nullptr, bufA, scaleB, shiftB, stream);
  run_mp<64, 64>(bufA, nn, etyp, r1m, bufC, stream);
  run_bn(bufC, 64, r1m.g, r1m.be, nullptr, bufC, scaleB, shiftB, stream);
  run_gemm(bufC, r1u.w, r1u.b, bufA, 64, 128, stream);
  run_bn(bufA, 128, r1u.g, r1u.be, bufB, bufA, scaleB, shiftB, stream);

  // --- conv2 + mp2 + conv3 ---
  run_gemm(bufA, c2.w, c2.b, bufB, 128, 128, stream);
  run_bn(bufB, 128, c2.g, c2.be, nullptr, bufB, scaleB, shiftB, stream);
  run_mp<128, 128>(bufB, nn, etyp, mp2, bufA, stream);
  run_bn(bufA, 128, mp2.g, mp2.be, nullptr, bufA, scaleB, shiftB, stream);
  run_gemm(bufA, c3.w, c3.b, bufB, 128, 256, stream);
  run_bn(bufB, 256, c3.g, c3.be, nullptr, bufB, scaleB, shiftB, stream);

  // --- res2 ---
  run_gemm(bufB, r2d.w, r2d.b, bufA, 256, 128, stream);
  run_bn(bufA, 128, r2d.g, r2d.be, nullptr, bufA, scaleB, shiftB, stream);
  run_mp<128, 128>(bufA, nn, etyp, r2m, bufC, stream);
  run_bn(bufC, 128, r2m.g, r2m.be, nullptr, bufC, scaleB, shiftB, stream);
  run_gemm(bufC, r2u.w, r2u.b, bufA, 128, 256, stream);
  run_bn(bufA, 256, r2u.g, r2u.be, bufB, bufA, scaleB, shiftB, stream);

  // --- conv4 + mp3 + conv5 ---
  run_gemm(bufA, c4.w, c4.b, bufB, 256, 256, stream);
  run_bn(bufB, 256, c4.g, c4.be, nullptr, bufB, scaleB, shiftB, stream);
  run_mp<256, 256>(bufB, nn, etyp, mp3, bufA, stream);
  run_bn(bufA, 256, mp3.g, mp3.be, nullptr, bufA, scaleB, shiftB, stream);
  run_gemm(bufA, c5.w, c5.b, bufB, 256, 512, stream);
  run_bn(bufB, 512, c5.g, c5.be, nullptr, bufB, scaleB, shiftB, stream);

  // --- res3 ---
  run_gemm(bufB, r3d.w, r3d.b, bufA, 512, 256, stream);
  run_bn(bufA, 256, r3d.g, r3d.be, nullptr, bufA, scaleB, shiftB, stream);
  run_mp<256, 256>(bufA, nn, etyp, r3m, bufC, stream);
  run_bn(bufC, 256, r3m.g, r3m.be, nullptr, bufC, scaleB, shiftB, stream);
  run_gemm(bufC, r3u.w, r3u.b, bufA, 256, 512, stream);
  run_bn(bufA, 512, r3u.g, r3u.be, bufB, bufA, scaleB, shiftB, stream);

  // --- affine + normalize + mask ---
  run_gemm(bufA, aff_w, nullptr, bufB, 512, 512, stream);
  norm_out_kernel<<<Mn, 256, 0, stream>>>(bufB, mask, (float*)d_out);
}